// metapath2vec_31129922962024
// MI455X (gfx1250) — compile-verified
//
#include <hip/hip_runtime.h>

// metapath2vec skip-gram loss, MI455X (gfx1250, wave32).
// Positives  : per-b 21x21 Gram matrix via V_WMMA_F32_16X16X4_F32 (f32 WMMA,
//              matches f32 reference numerics).
// Negatives  : gather-bound (≈485 MB of random 512B rows @ 23.3 TB/s HBM);
//              wave-per-dot coalesced loads + shfl_xor butterfly reduction,
//              with global_prefetch of the next row.
// Reduction  : per-b partials -> fixed-order tree (deterministic, no atomics).

#define NPAIRS 185
#define PLEN   21
#define KNEG   5
#define BATCH  1024
#define DIM    128
#define ESTR   132   // LDS row stride (floats): conflict-free WMMA frag loads

typedef float v2f __attribute__((ext_vector_type(2)));
typedef float v8f __attribute__((ext_vector_type(8)));

__device__ __forceinline__ float log_sigmoid(float x) {
    // stable: min(x,0) - log1p(exp(-|x|))
    return fminf(x, 0.0f) - log1pf(__expf(-fabsf(x)));
}

__global__ __launch_bounds__(256) void mp2v_main(
    const float* __restrict__ emb,
    const int*   __restrict__ path,
    const int*   __restrict__ neg,
    float*       __restrict__ partial)
{
    __shared__ float E[32 * ESTR];     // padded 32x128 path-embedding tile
    __shared__ float G[32 * 32];       // Gram matrix (only 21x21 valid)
    __shared__ int   pair_i[NPAIRS];
    __shared__ int   pair_j[NPAIRS];
    __shared__ float red[256];

    const int b    = blockIdx.x;
    const int tid  = threadIdx.x;
    const int lane = tid & 31;
    const int wave = tid >> 5;

    // Build the static skip-gram window pair table (cheap, one thread).
    if (tid == 255) {
        int t = 0;
        for (int i = 0; i < PLEN; ++i) {
            int j0 = i - KNEG; if (j0 < 0) j0 = 0;
            int j1 = i + KNEG; if (j1 > PLEN) j1 = PLEN;
            for (int j = j0; j < j1; ++j) { pair_i[t] = i; pair_j[t] = j; ++t; }
        }
    }

    // Cooperative gather of the 21 path embeddings (each loaded ONCE per b,
    // instead of 2x185 times as in the reference); rows 21..31 zero-padded.
    for (int s = tid; s < 32 * 32; s += 256) {
        const int row = s >> 5;          // 0..31
        const int c4  = s & 31;          // float4 column
        float4 v = make_float4(0.f, 0.f, 0.f, 0.f);
        if (row < PLEN) {
            const long r = (long)path[b * PLEN + row];
            v = ((const float4*)(emb + r * DIM))[c4];
        }
        *(float4*)(&E[row * ESTR + c4 * 4]) = v;   // 528B row stride, 16B aligned
    }
    __syncthreads();

    // ---- wave 0: G = E * E^T via WMMA f32 16x16x4 (K swept 128/4 = 32x) ----
    // Fragment layout (16x4 f32): lane m -> {K0,K1} of row m; lane 16+m ->
    // {K2,K3} of row m. For a Gram matrix the Ntile B-fragment of E^T has the
    // identical per-lane contents as the Mtile A-fragment, so one ds_load_b64
    // per tile per k-step feeds both operands.
    if (wave == 0) {
        v8f g00 = {}, g01 = {}, g10 = {}, g11 = {};
        const int mrow = lane & 15;
        const int koff = (lane >> 4) * 2;
        #pragma unroll 4
        for (int kk = 0; kk < 32; ++kk) {
            const int cb = kk * 4 + koff;
            v2f a0 = *(const v2f*)(&E[ mrow       * ESTR + cb]);
            v2f a1 = *(const v2f*)(&E[(16 + mrow) * ESTR + cb]);
            g00 = __builtin_amdgcn_wmma_f32_16x16x4_f32(false, a0, false, a0, (short)0, g00, false, false);
            g01 = __builtin_amdgcn_wmma_f32_16x16x4_f32(false, a0, false, a1, (short)0, g01, false, false);
            g10 = __builtin_amdgcn_wmma_f32_16x16x4_f32(false, a1, false, a0, (short)0, g10, false, false);
            g11 = __builtin_amdgcn_wmma_f32_16x16x4_f32(false, a1, false, a1, (short)0, g11, false, false);
        }
        // C/D layout: lane L VGPR v holds M = v + 8*(L>>4), N = L&15.
        const int n     = lane & 15;
        const int mbase = (lane >> 4) * 8;
        #pragma unroll
        for (int v = 0; v < 8; ++v) {
            const int m = mbase + v;
            G[ m       * 32 + n     ] = g00[v];
            G[ m       * 32 + 16 + n] = g01[v];
            G[(16 + m) * 32 + n     ] = g10[v];
            G[(16 + m) * 32 + 16 + n] = g11[v];
        }
    }

    // ---- negatives: wave-per-dot, 925 dots per b (waves 1..7 start now,
    //      wave 0 joins after the Gram). Each dot: coalesced 512B row load. ----
    float acc = 0.0f;
    for (int d = wave; d < NPAIRS * KNEG; d += 8) {
        const int p  = d / KNEG;
        const int k  = d - p * KNEG;
        const int qi = pair_i[p];
        const long idx = (long)neg[((long)p * BATCH + b) * KNEG + k];

        // prefetch next negative row for this wave
        const int dn = d + 8;
        if (dn < NPAIRS * KNEG) {
            const int pn = dn / KNEG;
            const int kn = dn - pn * KNEG;
            const long idxn = (long)neg[((long)pn * BATCH + b) * KNEG + kn];
            __builtin_prefetch((const void*)(emb + idxn * DIM + lane * 4), 0, 1);
        }

        const float4 nv = ((const float4*)(emb + idx * DIM))[lane];
        const float4 qv = *(const float4*)(&E[qi * ESTR + lane * 4]);
        float s = nv.x * qv.x + nv.y * qv.y + nv.z * qv.z + nv.w * qv.w;
        #pragma unroll
        for (int off = 16; off > 0; off >>= 1)
            s += __shfl_xor(s, off, 32);
        if (lane == 0) acc += log_sigmoid(-s);   // once per wave per dot
    }

    __syncthreads();   // G ready for everyone

    // ---- positives: read Gram entries through the window pair table ----
    for (int t = tid; t < NPAIRS; t += 256)
        acc += log_sigmoid(G[pair_i[t] * 32 + pair_j[t]]);

    // ---- deterministic block reduction -> partial[b] ----
    red[tid] = acc;
    __syncthreads();
    if (tid == 0) {
        float s = 0.0f;
        for (int t = 0; t < 256; ++t) s += red[t];
        partial[b] = s;
    }
}

__global__ __launch_bounds__(256) void mp2v_reduce(
    const float* __restrict__ partial,
    float*       __restrict__ out)
{
    __shared__ float red[256];
    float s = 0.0f;
    for (int t = threadIdx.x; t < BATCH; t += 256) s += partial[t];
    red[threadIdx.x] = s;
    __syncthreads();
    if (threadIdx.x == 0) {
        float tot = 0.0f;
        for (int t = 0; t < 256; ++t) tot += red[t];
        out[0] = -tot * (1.0f / (float)BATCH);
    }
}

extern "C" void kernel_launch(void* const* d_in, const int* in_sizes, int n_in,
                              void* d_out, int out_size, void* d_ws, size_t ws_size,
                              hipStream_t stream) {
    const float* emb  = (const float*)d_in[0];   // (500000,128) f32
    const int*   path = (const int*)  d_in[1];   // (1024,21)    i32
    const int*   neg  = (const int*)  d_in[2];   // (185,1024,5) i32
    float* partial = (float*)d_ws;               // 1024 floats of scratch
    float* out     = (float*)d_out;              // scalar f32

    mp2v_main  <<<BATCH, 256, 0, stream>>>(emb, path, neg, partial);
    mp2v_reduce<<<1,     256, 0, stream>>>(partial, out);
}